// GCN_85229331022439
// MI455X (gfx1250) — compile-verified
//
#include <hip/hip_runtime.h>

typedef __attribute__((ext_vector_type(2))) float v2f;
typedef __attribute__((ext_vector_type(8))) float v8f;

#define NNODES 50000
#define NEDGES 800000
#define NGRAPH 256
#define NCLS   16

// ---------------------------------------------------------------- zero ----
__global__ void zero_f32(float* __restrict__ p, long n) {
    long t = (long)blockIdx.x * blockDim.x + threadIdx.x;
    if (t < n) p[t] = 0.0f;
}

// ------------------------------------------------------- edge aggregate ----
// agg[dst[e]][:] += h[src[e]][:]   for 64-wide features.
// One thread per (edge, 4-feature chunk): float4 gather + 4 f32 atomics.
__global__ void aggregate64(const float* __restrict__ h,
                            const int* __restrict__ src,
                            const int* __restrict__ dst,
                            float* __restrict__ agg, int e) {
    long t = (long)blockIdx.x * blockDim.x + threadIdx.x;
    long total = (long)e * 16;
    if (t >= total) return;
    int edge = (int)(t >> 4);
    int c4   = (int)(t & 15) << 2;
    int s = src[edge];
    int d = dst[edge];
    const float4 v = *(const float4*)(h + (size_t)s * 64 + c4);
    float* p = agg + (size_t)d * 64 + c4;
    atomicAdd(p + 0, v.x);
    atomicAdd(p + 1, v.y);
    atomicAdd(p + 2, v.z);
    atomicAdd(p + 3, v.w);
}

// ------------------------------------------------ fused dual GEMM (WMMA) ----
// out[i] = act( h[i]@Wroot + agg[i]@Wrel + b )   with Din=64, DOUT in {64,16}.
// Fused as (N x 128) @ (128 x DOUT) : K-steps 0..15 read h/Wroot, 16..31 read
// agg/Wrel. One wave owns a 16 x DOUT output strip; 4 waves per block.
// V_WMMA_F32_16X16X4_F32 fragment layouts (ISA 7.12.2):
//   A 16x4 f32: lane<16 -> {K=k,k+1}, lane>=16 -> {K=k+2,k+3}, M = lane%16
//   B 4x16 f32: same K split per half-wave, N = lane%16
//   C/D 16x16:  VGPR r -> M = r + 8*(lane>=16), N = lane%16
template <int DOUT, bool RELU>
__global__ void gemm_dual_wmma(const float* __restrict__ h,
                               const float* __restrict__ agg,
                               const float* __restrict__ wroot,
                               const float* __restrict__ wrel,
                               const float* __restrict__ bias,
                               float* __restrict__ out, int n) {
    const int lane = threadIdx.x & 31;
    const int wave = threadIdx.x >> 5;
    const int m0   = blockIdx.x * 64 + wave * 16;
    const int hi   = lane >> 4;   // half-wave select (K offset +2)
    const int lm   = lane & 15;   // M for A-rows / N for B,C,D

    int arow = m0 + lm;
    if (arow >= n) arow = n - 1;  // clamp loads: EXEC must stay all-ones

    constexpr int NT = DOUT / 16;
    v8f acc[NT];
#pragma unroll
    for (int t = 0; t < NT; ++t) acc[t] = (v8f)0.0f;

#pragma unroll 4
    for (int ks = 0; ks < 32; ++ks) {
        const float* srcm = (ks < 16) ? h : agg;
        const float* W    = (ks < 16) ? wroot : wrel;
        const int k4   = (ks & 15) * 4;
        const int krow = k4 + 2 * hi;

        v2f a;
        const float* ap = srcm + (size_t)arow * 64 + krow;
        a.x = ap[0];
        a.y = ap[1];

#pragma unroll
        for (int t = 0; t < NT; ++t) {
            v2f b;
            b.x = W[(size_t)krow * DOUT + t * 16 + lm];
            b.y = W[(size_t)(krow + 1) * DOUT + t * 16 + lm];
            acc[t] = __builtin_amdgcn_wmma_f32_16x16x4_f32(
                false, a, false, b, (short)0, acc[t], false, false);
        }
    }

#pragma unroll
    for (int t = 0; t < NT; ++t) {
        const float bc = bias[t * 16 + lm];
#pragma unroll
        for (int r = 0; r < 8; ++r) {
            int orow = m0 + r + 8 * hi;
            if (orow < n) {
                float v = acc[t][r] + bc;
                if (RELU) v = fmaxf(v, 0.0f);
                out[(size_t)orow * DOUT + t * 16 + lm] = v;
            }
        }
    }
}

// ------------------------------------------------------------- mean pool ----
__global__ void pool_sum(const float* __restrict__ h3,
                         const int* __restrict__ batch,
                         float* __restrict__ sums,
                         float* __restrict__ counts, int n) {
    long t = (long)blockIdx.x * blockDim.x + threadIdx.x;
    if (t >= (long)n * NCLS) return;
    int i = (int)(t >> 4);
    int c = (int)(t & 15);
    int g = batch[i];
    atomicAdd(&sums[g * NCLS + c], h3[(size_t)i * NCLS + c]);
    if (c == 0) atomicAdd(&counts[g], 1.0f);
}

__global__ void pool_finalize(const float* __restrict__ sums,
                              const float* __restrict__ counts,
                              float* __restrict__ out) {
    int t = blockIdx.x * blockDim.x + threadIdx.x;
    if (t >= NGRAPH * NCLS) return;
    int g = t >> 4;
    out[t] = sums[t] / fmaxf(counts[g], 1.0f);
}

// ---------------------------------------------------------------- launch ----
extern "C" void kernel_launch(void* const* d_in, const int* in_sizes, int n_in,
                              void* d_out, int out_size, void* d_ws, size_t ws_size,
                              hipStream_t stream) {
    const float* x       = (const float*)d_in[0];
    const int*   eidx    = (const int*)d_in[1];   // (2, E): row0=src, row1=dst
    const int*   batch   = (const int*)d_in[2];
    const float* w1_root = (const float*)d_in[3];
    const float* w1_rel  = (const float*)d_in[4];
    const float* b1      = (const float*)d_in[5];
    const float* w2_root = (const float*)d_in[6];
    const float* w2_rel  = (const float*)d_in[7];
    const float* b2      = (const float*)d_in[8];
    const float* w3_root = (const float*)d_in[9];
    const float* w3_rel  = (const float*)d_in[10];
    const float* b3      = (const float*)d_in[11];
    float* out = (float*)d_out;

    const int E = in_sizes[1] / 2;
    const int N = in_sizes[2];
    const int* src = eidx;
    const int* dst = eidx + E;

    const long nf = (long)N * 64;
    float* buf0   = (float*)d_ws;          // agg scratch (N x 64)
    float* buf1   = buf0 + nf;             // h1 (N x 64), later h3 (N x 16)
    float* buf2   = buf1 + nf;             // h2 (N x 64)
    float* sums   = buf2 + nf;             // (NGRAPH x NCLS)
    float* counts = sums + NGRAPH * NCLS;  // (NGRAPH)

    const int ZB = 256;
    const long aggWork  = (long)E * 16;
    const int  aggGrid  = (int)((aggWork + ZB - 1) / ZB);
    const int  gemmGrid = (N + 63) / 64;   // 4 waves x 16 rows per block
    const long poolWork = (long)N * NCLS;

    // ---- layer 1: agg(x) ; h1 = relu(x@W1r + agg@W1l + b1)
    zero_f32<<<(int)((nf + ZB - 1) / ZB), ZB, 0, stream>>>(buf0, nf);
    aggregate64<<<aggGrid, ZB, 0, stream>>>(x, src, dst, buf0, E);
    gemm_dual_wmma<64, true><<<gemmGrid, 128, 0, stream>>>(
        x, buf0, w1_root, w1_rel, b1, buf1, N);

    // ---- layer 2: agg(h1) ; h2 = relu(h1@W2r + agg@W2l + b2)
    zero_f32<<<(int)((nf + ZB - 1) / ZB), ZB, 0, stream>>>(buf0, nf);
    aggregate64<<<aggGrid, ZB, 0, stream>>>(buf1, src, dst, buf0, E);
    gemm_dual_wmma<64, true><<<gemmGrid, 128, 0, stream>>>(
        buf1, buf0, w2_root, w2_rel, b2, buf2, N);

    // ---- layer 3: agg(h2) ; h3 = h2@W3r + agg@W3l + b3   (no relu)
    zero_f32<<<(int)((nf + ZB - 1) / ZB), ZB, 0, stream>>>(buf0, nf);
    aggregate64<<<aggGrid, ZB, 0, stream>>>(buf2, src, dst, buf0, E);
    gemm_dual_wmma<16, false><<<gemmGrid, 128, 0, stream>>>(
        buf2, buf0, w3_root, w3_rel, b3, buf1, N);

    // ---- mean pool over graph ids
    const long pz = NGRAPH * NCLS + NGRAPH;
    zero_f32<<<(int)((pz + ZB - 1) / ZB), ZB, 0, stream>>>(sums, pz);
    pool_sum<<<(int)((poolWork + ZB - 1) / ZB), ZB, 0, stream>>>(
        buf1, batch, sums, counts, N);
    pool_finalize<<<(NGRAPH * NCLS + ZB - 1) / ZB, ZB, 0, stream>>>(
        sums, counts, out);
}